// CRF_8950711845018
// MI455X (gfx1250) — compile-verified
//
#include <hip/hip_runtime.h>
#include <hip/hip_bf16.h>
#include <stdint.h>

// Problem constants (fixed by the reference file)
#define TAG      50      // TARGET_SIZE + 2
#define TAGP     52      // padded row stride: 52 floats = 13 x 16B, keeps b128 loads aligned
#define SEQ      512
#define TSTART   48
#define TEND     49
#define NTHREADS 64

typedef unsigned int u32x4 __attribute__((ext_vector_type(4)));
typedef int          i32x8 __attribute__((ext_vector_type(8)));
typedef int          i32x4 __attribute__((ext_vector_type(4)));

__global__ __launch_bounds__(NTHREADS)
void viterbi_crf_kernel(const float* __restrict__ feats,
                        const unsigned char* __restrict__ mask,
                        const float* __restrict__ trans,
                        float* __restrict__ out)
{
    __shared__ __align__(16) float s_trans[TAG * TAG];   // 10 KB
    __shared__ __align__(16) float s_part[2][TAGP];      // double-buffered partition (16B-aligned rows)
    __shared__ float               s_last[TAG];          // partition at last_pos
    __shared__ unsigned char       s_bp[(SEQ - 1) * TAG];// 25 KB backpointers (LDS-resident backtrack)
    __shared__ int                 s_red[NTHREADS];
    __shared__ int                 s_len;

    const int b   = blockIdx.x;
    const int tid = threadIdx.x;
    const float*         fb = feats + (size_t)b * SEQ * TAG;
    const unsigned char* mb = mask  + (size_t)b * SEQ;

    // ---- Stage the 50x50 transition matrix into LDS via the Tensor Data Mover ----
#if defined(__gfx1250__) && __has_builtin(__builtin_amdgcn_tensor_load_to_lds) && __has_builtin(__builtin_amdgcn_s_wait_tensorcnt)
    {
        const uint64_t ga  = (uint64_t)(uintptr_t)trans;
        const uint32_t lds = (uint32_t)(uintptr_t)&s_trans[0]; // LDS aperture: addr[31:0] = LDS byte offset
        u32x4 g0;
        g0[0] = 1u;                                            // count=1 (valid user descriptor)
        g0[1] = lds;                                           // lds_addr [63:32]
        g0[2] = (uint32_t)ga;                                  // global_addr [95:64]
        g0[3] = (uint32_t)((ga >> 32) & 0x01FFFFFFu)           // global_addr [120:96]
              | (2u << 30);                                    // type=2 ("image")
        i32x8 g1;
        g1[0] = (int)(2u << 16);                               // data_size=2 -> 4 bytes; wg_mask=0 (not in cluster)
        g1[1] = (int)((uint32_t)TAG << 16);                    // tensor_dim0 = 50   (bits 79:48, low half)
        g1[2] = (int)((uint32_t)TAG << 16);                    // tensor_dim1 = 50   (bits 111:80, low half)
        g1[3] = (int)((uint32_t)TAG << 16);                    // tile_dim0   = 50   (bits 127:112)
        g1[4] = (int)TAG;                                      // tile_dim1   = 50   (bits 143:128)
        g1[5] = (int)TAG;                                      // tensor_dim0_stride = 50 (bits 191:160)
        g1[6] = 0; g1[7] = 0;
        i32x4 z4 = {0, 0, 0, 0};                               // groups 2/3 unused (2D tensor)
        i32x8 z8 = {0, 0, 0, 0, 0, 0, 0, 0};                   // extra group (clang-23 6-arg form): inert zeros
        __builtin_amdgcn_tensor_load_to_lds(g0, g1, z4, z4, z8, 0);
        __builtin_amdgcn_s_wait_tensorcnt(0);
    }
#else
    for (int k = tid; k < TAG * TAG; k += NTHREADS) s_trans[k] = trans[k];
#endif

    // ---- Sequence length = sum of mask row (mask is a prefix mask: arange < length) ----
    int acc = 0;
    for (int k = tid; k < SEQ; k += NTHREADS) acc += (int)mb[k];
    s_red[tid] = acc;
    __syncthreads();
    if (tid == 0) {
        int L = 0;
        for (int k = 0; k < NTHREADS; ++k) L += s_red[k];
        s_len = L;
    }
    __syncthreads();
    const int length   = s_len;
    const int last_pos = length - 1;  // >= S/2 - 1 = 255

    // ---- Per-thread register cache of transition column j: trj[i] = trans[i][j] ----
    const int j = tid;
    float trj[TAG];
    if (j < TAG) {
#pragma unroll
        for (int i = 0; i < TAG; ++i) trj[i] = s_trans[i * TAG + j];
        // part0[j] = feats[b,0,j] + trans[START,j]
        const float p0 = fb[j] + s_trans[TSTART * TAG + j];
        s_part[0][j] = p0;
        s_last[j]    = p0;
    }
    __syncthreads();

    // ---- Forward recursion: 511 steps, one barrier per step ----
    for (int t = 1; t < SEQ; ++t) {
        const int rb = (t - 1) & 1;
        const int wb = t & 1;
        if (j < TAG) {
            int tp = t + 8; if (tp > SEQ - 1) tp = SEQ - 1;
            __builtin_prefetch(&fb[tp * TAG + j], 0, 1);       // -> global_prefetch_b8

            const float f = fb[t * TAG + j];
            float bestV = -3.0e38f;
            int   bestI = 0;
#pragma unroll
            for (int i = 0; i < TAG; ++i) {
                const float cand = s_part[rb][i] + trj[i];     // broadcast ds_load_b128 + register column
                if (cand > bestV) { bestV = cand; bestI = i; } // strict '>' == first-occurrence argmax
            }
            const float np = bestV + f;                        // partition updates unconditionally
            s_part[wb][j] = np;
            // mask is a prefix mask, so m_t == (t < length); no global load in the hot loop
            s_bp[(t - 1) * TAG + j] = (t < length) ? (unsigned char)bestI : (unsigned char)0;
            if (t == last_pos) s_last[j] = np;                 // part_hist[last_pos]
        }
        __syncthreads();
    }

    // ---- Terminal argmax + sequential backtrack (thread 0; all LDS-resident) ----
    if (tid == 0) {
        float bestV = -3.0e38f;
        int   ptr   = 0;
        for (int i = 0; i < TAG; ++i) {
            const float v = s_last[i] + s_trans[i * TAG + TEND];
            if (v > bestV) { bestV = v; ptr = i; }
        }
        const int pointer = ptr;
        float* ob = out + (size_t)b * SEQ;
        ob[SEQ - 1] = (float)pointer;
        for (int i2 = SEQ - 2; i2 >= 0; --i2) {
            // back_points[last_pos] was overwritten with `pointer` in the reference
            ptr = (i2 == last_pos) ? pointer : (int)s_bp[i2 * TAG + ptr];
            ob[i2] = (float)ptr;
        }
    }
}

extern "C" void kernel_launch(void* const* d_in, const int* in_sizes, int n_in,
                              void* d_out, int out_size, void* d_ws, size_t ws_size,
                              hipStream_t stream)
{
    const float*         feats = (const float*)d_in[0];          // (B, S, TAG) f32
    const unsigned char* mask  = (const unsigned char*)d_in[1];  // (B, S) bool (1 byte)
    const float*         trans = (const float*)d_in[2];          // (TAG, TAG) f32
    float*               out   = (float*)d_out;                  // (B, S) path indices as f32

    const int B = in_sizes[1] / SEQ;                             // 128 for the reference setup
    viterbi_crf_kernel<<<dim3(B), dim3(NTHREADS), 0, stream>>>(feats, mask, trans, out);
}